// GCNBlock_20744692039840
// MI455X (gfx1250) — compile-verified
//
#include <hip/hip_runtime.h>
#include <hip/hip_bf16.h>

typedef float v2f __attribute__((ext_vector_type(2)));
typedef float v8f __attribute__((ext_vector_type(8)));

#define HDIM 128
#define NGRAPH 64
#define GN_EPS 1e-5f

// ---------------------------------------------------------------------------
// 1) init: deg = 1.0 (self-loop weight), zero cnt/gsum/gvar and flat output
// ---------------------------------------------------------------------------
__global__ __launch_bounds__(256) void k_init(float* __restrict__ deg,
                                              float* __restrict__ cnt,
                                              float* __restrict__ gsum,
                                              float* __restrict__ gvar,
                                              float* __restrict__ flat,
                                              int N) {
  int idx = blockIdx.x * blockDim.x + threadIdx.x;
  int stride = gridDim.x * blockDim.x;
  for (int i = idx; i < N; i += stride) deg[i] = 1.0f;
  for (int i = idx; i < NGRAPH; i += stride) cnt[i] = 0.0f;
  for (int i = idx; i < NGRAPH * HDIM; i += stride) {
    gsum[i] = 0.0f;
    gvar[i] = 0.0f;
    flat[i] = 0.0f;  // ReLU output >= 0, so +0.0 is the identity for max
  }
}

// ---------------------------------------------------------------------------
// 2) weighted in-degree: deg[col[e]] += w[e]
// ---------------------------------------------------------------------------
__global__ __launch_bounds__(256) void k_deg(const int* __restrict__ col,
                                             const float* __restrict__ w,
                                             float* __restrict__ deg, int E) {
  int e = blockIdx.x * blockDim.x + threadIdx.x;
  if (e < E) atomicAdd(&deg[col[e]], w[e]);
}

// ---------------------------------------------------------------------------
// 3) dis = deg > 0 ? rsqrt(deg) : 0  (in place)
// ---------------------------------------------------------------------------
__global__ __launch_bounds__(256) void k_dis(float* __restrict__ deg, int N) {
  int i = blockIdx.x * blockDim.x + threadIdx.x;
  if (i < N) {
    float d = deg[i];
    deg[i] = (d > 0.0f) ? rsqrtf(d) : 0.0f;
  }
}

// ---------------------------------------------------------------------------
// 4) x = inputs @ W via V_WMMA_F32_16X16X4_F32 (fp32-exact tensor path).
//    8 waves/block, wave w owns output cols [16w,16w+16); B (W slice) lives in
//    64 VGPRs/lane for the whole kernel. A tiles are DOUBLE-BUFFERED in LDS
//    and staged with GLOBAL_LOAD_ASYNC_TO_LDS_B128 (ASYNCcnt-tracked), so the
//    async engine fills buffer nxt while the WMMA chain runs on buffer cur.
//    Pitch 132 floats (528 B = 33*16) keeps 16B alignment and spreads the
//    half-wave ds_load_b64 reads across 16 banks.
// ---------------------------------------------------------------------------
__global__ __launch_bounds__(256) void k_gemm_wmma(const float* __restrict__ A,
                                                   const float* __restrict__ W,
                                                   float* __restrict__ X,
                                                   int N, int numTiles) {
  __shared__ __align__(16) float As[2][16 * 132];
  const int lane = threadIdx.x & 31;
  const int wave = threadIdx.x >> 5;          // 0..7 -> 16-col tile
  const int k0 = (lane >> 4) * 2;             // K sub-offset per half-wave
  const int nloc = lane & 15;                 // output column within tile
  const int ncol = wave * 16 + nloc;          // global output column

  // Resident B: lane holds W[4kb+k0+j][ncol], j in {0,1}, kb = 0..31
  v2f breg[32];
#pragma unroll
  for (int kb = 0; kb < 32; ++kb) {
    int k = kb * 4 + k0;
    v2f t;
    t.x = W[k * HDIM + ncol];
    t.y = W[(k + 1) * HDIM + ncol];
    breg[kb] = t;
  }

  // issue async memory->LDS copies for one 16x128 tile into buffer `buf`
  auto issue_tile = [&](int tile, int buf) {
#pragma unroll
    for (int t = 0; t < 2; ++t) {
      int idx = threadIdx.x + t * 256;   // 0..511 float4 slots
      int r = idx >> 5;                  // row in tile (32 float4 per row)
      int c4 = idx & 31;                 // float4 column
      int grow = tile * 16 + r;
      if (grow > N - 1) grow = N - 1;    // clamp: extra rows never stored
      unsigned long long gaddr =
          (unsigned long long)(const void*)(A + (size_t)grow * HDIM + c4 * 4);
      // low 32 bits of a flat pointer to __shared__ == LDS byte offset
      unsigned ldsoff =
          (unsigned)(unsigned long long)(const void*)&As[buf][r * 132 + c4 * 4];
      asm volatile("global_load_async_to_lds_b128 %0, %1, off"
                   :: "v"(ldsoff), "v"(gaddr)
                   : "memory");
    }
  };

  int tile = blockIdx.x;
  if (tile < numTiles) issue_tile(tile, 0);

  int buf = 0;
  for (; tile < numTiles; tile += gridDim.x, buf ^= 1) {
    // my async stores into As[buf] have landed in LDS
    asm volatile("s_wait_asynccnt 0x0" ::: "memory");
    // everyone's stores landed AND everyone finished reading As[buf^1]
    __syncthreads();

    int ntile = tile + gridDim.x;
    if (ntile < numTiles) issue_tile(ntile, buf ^ 1);  // overlap with compute

    v8f acc = {0.f, 0.f, 0.f, 0.f, 0.f, 0.f, 0.f, 0.f};
    const int m = lane & 15;
#pragma unroll
    for (int kb = 0; kb < 32; ++kb) {
      int k = kb * 4 + k0;
      v2f a = *(const v2f*)&As[buf][m * 132 + k];   // 8B-aligned ds_load_b64
      acc = __builtin_amdgcn_wmma_f32_16x16x4_f32(
          false, a, false, breg[kb], (short)0, acc, false, false);
    }

    // C/D layout: VGPR r -> row (r + 8*(lane>>4)), col = lane&15
    const int row0 = tile * 16;
    const int mbase = (lane >> 4) * 8;
#pragma unroll
    for (int r = 0; r < 8; ++r) {
      int grow = row0 + mbase + r;
      if (grow < N) X[(size_t)grow * HDIM + ncol] = acc[r];
    }
  }
}

// ---------------------------------------------------------------------------
// 5) out[i] = x[i] * (dis[i]^2) + b   (self-loop term + bias); also count cnt.
//    One wave per node; lane -> float4 feature chunk (32*4 = 128).
// ---------------------------------------------------------------------------
__global__ __launch_bounds__(256) void k_init_out(const float* __restrict__ x,
                                                  const float* __restrict__ dis,
                                                  const float* __restrict__ bias,
                                                  const int* __restrict__ batch,
                                                  float* __restrict__ out,
                                                  float* __restrict__ cnt,
                                                  int N) {
  int i = blockIdx.x * 8 + (threadIdx.x >> 5);
  if (i >= N) return;
  int lane = threadIdx.x & 31;
  float d = dis[i];
  float sn = d * d;  // dis[i] * 1.0 * dis[i]
  float4 xv = ((const float4*)(x + (size_t)i * HDIM))[lane];
  float4 bv = ((const float4*)bias)[lane];
  float4 o;
  o.x = xv.x * sn + bv.x;
  o.y = xv.y * sn + bv.y;
  o.z = xv.z * sn + bv.z;
  o.w = xv.w * sn + bv.w;
  ((float4*)(out + (size_t)i * HDIM))[lane] = o;
  if (lane == 0) atomicAdd(&cnt[batch[i]], 1.0f);
}

// ---------------------------------------------------------------------------
// 6) edge scatter: out[col[e]] += x[row[e]] * (dis[row]*w*dis[col]).
//    One wave per edge; lane loads one float4 (global_load_b128) and does 4
//    global_atomic_add_f32.
// ---------------------------------------------------------------------------
__global__ __launch_bounds__(256) void k_edges(const int* __restrict__ row,
                                               const int* __restrict__ col,
                                               const float* __restrict__ ew,
                                               const float* __restrict__ dis,
                                               const float* __restrict__ x,
                                               float* __restrict__ out, int E) {
  int e = blockIdx.x * 8 + (threadIdx.x >> 5);
  if (e >= E) return;
  int lane = threadIdx.x & 31;
  int r = row[e];
  int c = col[e];
  float nrm = dis[r] * ew[e] * dis[c];
  float4 xv = ((const float4*)(x + (size_t)r * HDIM))[lane];
  float* dst = out + (size_t)c * HDIM + lane * 4;
  atomicAdd(dst + 0, xv.x * nrm);
  atomicAdd(dst + 1, xv.y * nrm);
  atomicAdd(dst + 2, xv.z * nrm);
  atomicAdd(dst + 3, xv.w * nrm);
}

// ---------------------------------------------------------------------------
// 7) per-graph feature sums (batch is sorted: accumulate in registers, flush
//    one atomic per graph boundary per thread).
// ---------------------------------------------------------------------------
__global__ __launch_bounds__(128) void k_gsum(const float* __restrict__ out,
                                              const int* __restrict__ batch,
                                              float* __restrict__ gsum,
                                              int N, int npb) {
  int h = threadIdx.x;
  int i0 = blockIdx.x * npb;
  int i1 = min(i0 + npb, N);
  float acc = 0.f;
  int gcur = -1;
  for (int i = i0; i < i1; ++i) {
    int g = batch[i];
    if (g != gcur) {
      if (gcur >= 0) atomicAdd(&gsum[gcur * HDIM + h], acc);
      gcur = g;
      acc = 0.f;
    }
    acc += out[(size_t)i * HDIM + h];
  }
  if (gcur >= 0) atomicAdd(&gsum[gcur * HDIM + h], acc);
}

// ---------------------------------------------------------------------------
// 8) per-graph variance of (out - alpha*mean)
// ---------------------------------------------------------------------------
__global__ __launch_bounds__(128) void k_gvar(const float* __restrict__ out,
                                              const int* __restrict__ batch,
                                              const float* __restrict__ gsum,
                                              const float* __restrict__ cnt,
                                              const float* __restrict__ alpha,
                                              float* __restrict__ gvar,
                                              int N, int npb) {
  int h = threadIdx.x;
  float al = alpha[h];
  int i0 = blockIdx.x * npb;
  int i1 = min(i0 + npb, N);
  float acc = 0.f;
  float am = 0.f;
  int gcur = -1;
  for (int i = i0; i < i1; ++i) {
    int g = batch[i];
    if (g != gcur) {
      if (gcur >= 0) atomicAdd(&gvar[gcur * HDIM + h], acc);
      gcur = g;
      float c = fmaxf(cnt[g], 1.0f);
      am = al * (gsum[g * HDIM + h] / c);
      acc = 0.f;
    }
    float s = out[(size_t)i * HDIM + h] - am;
    acc += s * s;
  }
  if (gcur >= 0) atomicAdd(&gvar[gcur * HDIM + h], acc);
}

// ---------------------------------------------------------------------------
// 9) normalize + ReLU (in place in d_out) + global max pool into flat.
//    ReLU output >= 0, so unsigned-bit atomicMax is an exact float max.
// ---------------------------------------------------------------------------
__global__ __launch_bounds__(256) void k_norm(const int* __restrict__ batch,
                                              const float* __restrict__ gsum,
                                              const float* __restrict__ gvar,
                                              const float* __restrict__ cnt,
                                              const float* __restrict__ alpha,
                                              const float* __restrict__ gamma,
                                              const float* __restrict__ beta,
                                              float* __restrict__ hbuf,
                                              float* __restrict__ flat,
                                              int N) {
  long long idx = (long long)blockIdx.x * blockDim.x + threadIdx.x;
  long long total = (long long)N * HDIM;
  if (idx >= total) return;
  int i = (int)(idx >> 7);
  int h = (int)(idx & (HDIM - 1));
  int g = batch[i];
  float c = fmaxf(cnt[g], 1.0f);
  float mean = gsum[g * HDIM + h] / c;
  float var = gvar[g * HDIM + h] / c;
  float sub = hbuf[idx] - alpha[h] * mean;
  float v = sub * rsqrtf(var + GN_EPS) * gamma[h] + beta[h];
  v = fmaxf(v, 0.0f);
  hbuf[idx] = v;
  atomicMax((unsigned int*)&flat[g * HDIM + h], __float_as_uint(v));
}

// ---------------------------------------------------------------------------
extern "C" void kernel_launch(void* const* d_in, const int* in_sizes, int n_in,
                              void* d_out, int out_size, void* d_ws, size_t ws_size,
                              hipStream_t stream) {
  const float* inputs = (const float*)d_in[0];
  const int* edge_index = (const int*)d_in[1];
  const int* batch = (const int*)d_in[2];
  const float* edge_weight = (const float*)d_in[3];
  const float* W = (const float*)d_in[4];
  const float* bias = (const float*)d_in[5];
  const float* gn_weight = (const float*)d_in[6];
  const float* gn_bias = (const float*)d_in[7];
  const float* gn_alpha = (const float*)d_in[8];

  const int N = in_sizes[0] / HDIM;
  const int E = in_sizes[3];
  const int* erow = edge_index;       // edge_index[0]
  const int* ecol = edge_index + E;   // edge_index[1]

  // workspace layout (floats)
  float* ws = (float*)d_ws;
  float* x = ws;                                   // [N, 128]
  float* dis = x + (size_t)N * HDIM;               // [N] (deg -> dis in place)
  float* cnt = dis + N;                            // [64]
  float* gsum = cnt + NGRAPH;                      // [64,128]
  float* gvar = gsum + NGRAPH * HDIM;              // [64,128]

  float* hout = (float*)d_out;                     // [N,128]
  float* flat = hout + (size_t)N * HDIM;           // [64,128]

  const int numTiles = (N + 15) / 16;

  k_init<<<(N + 255) / 256, 256, 0, stream>>>(dis, cnt, gsum, gvar, flat, N);
  k_deg<<<(E + 255) / 256, 256, 0, stream>>>(ecol, edge_weight, dis, E);
  k_dis<<<(N + 255) / 256, 256, 0, stream>>>(dis, N);
  k_gemm_wmma<<<512, 256, 0, stream>>>(inputs, W, x, N, numTiles);
  k_init_out<<<(N + 7) / 8, 256, 0, stream>>>(x, dis, bias, batch, hout, cnt, N);
  k_edges<<<(E + 7) / 8, 256, 0, stream>>>(erow, ecol, edge_weight, dis, x, hout, E);
  const int nblk = 1024;
  const int npb = (N + nblk - 1) / nblk;
  k_gsum<<<nblk, 128, 0, stream>>>(hout, batch, gsum, N, npb);
  k_gvar<<<nblk, 128, 0, stream>>>(hout, batch, gsum, cnt, gn_alpha, gvar, N, npb);
  long long total = (long long)N * HDIM;
  k_norm<<<(int)((total + 255) / 256), 256, 0, stream>>>(
      batch, gsum, gvar, cnt, gn_alpha, gn_weight, gn_bias, hout, flat, N);
}